// VectorQuantizerEMA2D_54820962566231
// MI455X (gfx1250) — compile-verified
//
#include <hip/hip_runtime.h>
#include <math.h>

typedef __attribute__((ext_vector_type(2))) float v2f;
typedef __attribute__((ext_vector_type(4))) float v4f;
typedef __attribute__((ext_vector_type(8))) float v8f;

#define NCODES 512
#define DIMS   64
#define NROWS  131072          // 32 * 16 * 256
#define HT     4096            // 16 * 256 (ht index within one batch image)
#define SR     68              // padded LDS row stride (floats) -> conflict-free banks
#define CODES_PER_PASS 256     // two passes halve LDS -> 4 WGs / WGP

// d_out layout (flat float, reference tuple order)
#define OUT_ZQ     0
#define OUT_COMMIT 8388608
#define OUT_IDX    8388609     // 131072 indices stored as float
#define OUT_PPL    8519681
#define OUT_USAGE  8519682

__global__ void vq_init(float* __restrict__ out, float* __restrict__ counts) {
    int t = blockIdx.x * blockDim.x + threadIdx.x;
    if (t < NCODES) counts[t] = 0.0f;
    if (t == 0) out[OUT_COMMIT] = 0.0f;
}

__global__ __launch_bounds__(256)
void vq_main(const float* __restrict__ z_e, const float* __restrict__ embed,
             float* __restrict__ out, float* __restrict__ counts) {
    extern __shared__ float smem[];
    float* s_embed = smem;                                         // 256 * 68
    float* s_esq   = smem + CODES_PER_PASS * SR;                   // 256
    int*   s_widx  = (int*)(smem + CODES_PER_PASS * SR + CODES_PER_PASS); // 8*16

    const int t = threadIdx.x;

    const int wave  = t >> 5;
    const int lane  = t & 31;
    const int lhalf = lane & 15;
    const int dofs  = (lane < 16) ? 0 : 2;   // fp32 16x4 A layout: K pair (0,1 / 2,3) per half-wave

    const int n  = blockIdx.x * 128 + wave * 16 + lhalf;   // this lane's A row
    const int bI = n >> 12;
    const int ht = n & (HT - 1);
    const float* zrow = z_e + (size_t)bI * (DIMS * HT) + ht;

    // ---- A fragments: 16 chunks, each lane holds features (4c+dofs, 4c+dofs+1) ----
    v2f a[16];
    #pragma unroll
    for (int c = 0; c < 16; ++c) {
        a[c].x = zrow[(size_t)(4 * c + dofs) * HT];
        a[c].y = zrow[(size_t)(4 * c + dofs + 1) * HT];
    }

    float best[8];
    int   bidx[8];
    #pragma unroll
    for (int j = 0; j < 8; ++j) { best[j] = 3.4e38f; bidx[j] = 0; }

    // ---- two passes over the codebook: 256 codes staged in LDS per pass ----
    for (int p = 0; p < 2; ++p) {
        if (p > 0) __syncthreads();   // previous pass fully consumed before restage

        // stage: one codebook row per thread, padded stride + ||e||^2
        {
            const float* g = embed + (p * CODES_PER_PASS + t) * DIMS;
            float acc = 0.0f;
            #pragma unroll
            for (int j = 0; j < 16; ++j) {
                v4f v = *(const v4f*)(g + j * 4);
                *(v4f*)(s_embed + t * SR + j * 4) = v;
                acc += v.x * v.x + v.y * v.y + v.z * v.z + v.w * v.w;
            }
            s_esq[t] = acc;
        }
        __syncthreads();

        // ---- sweep 16 code tiles of 16 codes ----
        for (int ct = 0; ct < 16; ++ct) {
            const int codebase = ct * 16;                 // local (in-pass) code base
            const float* bcol = s_embed + (codebase + lhalf) * SR + dofs;
            v8f acc0 = {0.f, 0.f, 0.f, 0.f, 0.f, 0.f, 0.f, 0.f};
            v8f acc1 = {0.f, 0.f, 0.f, 0.f, 0.f, 0.f, 0.f, 0.f};
            #pragma unroll
            for (int c = 0; c < 16; c += 2) {
                v2f b0, b1;
                b0.x = bcol[4 * c];
                b0.y = bcol[4 * c + 1];
                b1.x = bcol[4 * (c + 1)];
                b1.y = bcol[4 * (c + 1) + 1];
                acc0 = __builtin_amdgcn_wmma_f32_16x16x4_f32(false, a[c], false, b0,
                                                             (short)0, acc0, false, false);
                acc1 = __builtin_amdgcn_wmma_f32_16x16x4_f32(false, a[c + 1], false, b1,
                                                             (short)0, acc1, false, false);
            }
            const float esq  = s_esq[codebase + lhalf];
            const int   code = p * CODES_PER_PASS + codebase + lhalf;
            #pragma unroll
            for (int j = 0; j < 8; ++j) {
                float dist = esq - 2.0f * (acc0[j] + acc1[j]);
                if (dist < best[j]) { best[j] = dist; bidx[j] = code; }
            }
        }
    }

    // ---- argmin across the 16 lanes of each half-wave group ----
    #pragma unroll
    for (int m = 1; m < 16; m <<= 1) {
        #pragma unroll
        for (int j = 0; j < 8; ++j) {
            float ov = __shfl_xor(best[j], m, 32);
            int   oi = __shfl_xor(bidx[j], m, 32);
            if (ov < best[j] || (ov == best[j] && oi < bidx[j])) {
                best[j] = ov; bidx[j] = oi;
            }
        }
    }

    // lanes 0-15 hold rows 0..7 (VGPR j), lanes 16-31 hold rows 8..15
    if (lane == 0) {
        #pragma unroll
        for (int j = 0; j < 8; ++j) s_widx[wave * 16 + j] = bidx[j];
    }
    if (lane == 16) {
        #pragma unroll
        for (int j = 0; j < 8; ++j) s_widx[wave * 16 + 8 + j] = bidx[j];
    }
    __syncthreads();

    const int widx = s_widx[wave * 16 + lhalf];

    // ---- z_q_st (== z_q forward value) + commitment partial from resident A regs.
    //      Winning row gathered from global (L2-hot); LDS only holds the last half. ----
    const float* erow  = embed + widx * DIMS + dofs;
    float*       obase = out + (size_t)bI * (DIMS * HT) + ht;
    float csum = 0.0f;
    #pragma unroll
    for (int c = 0; c < 16; ++c) {
        float q0 = erow[4 * c];
        float q1 = erow[4 * c + 1];
        obase[(size_t)(4 * c + dofs) * HT]     = q0;
        obase[(size_t)(4 * c + dofs + 1) * HT] = q1;
        float d0 = a[c].x - q0;
        float d1 = a[c].y - q1;
        csum += d0 * d0 + d1 * d1;
    }

    if (lane < 16) {
        out[OUT_IDX + n] = (float)widx;
        atomicAdd(&counts[widx], 1.0f);
    }

    // commitment: full-wave reduce then one atomic per wave
    #pragma unroll
    for (int m = 16; m >= 1; m >>= 1) csum += __shfl_xor(csum, m, 32);
    if (lane == 0) atomicAdd(&out[OUT_COMMIT], csum * (0.25f / 8388608.0f));
}

__global__ void vq_finalize(const float* __restrict__ counts, float* __restrict__ out) {
    __shared__ float sh[256];
    __shared__ float su[256];
    const int t = threadIdx.x;
    float h = 0.0f, u = 0.0f;
    for (int k = t; k < NCODES; k += 256) {
        float p = counts[k] * (1.0f / (float)NROWS);
        u += (p > 0.0f) ? 1.0f : 0.0f;
        h += -p * logf(p + 1e-12f);
    }
    sh[t] = h; su[t] = u;
    __syncthreads();
    for (int s = 128; s > 0; s >>= 1) {
        if (t < s) { sh[t] += sh[t + s]; su[t] += su[t + s]; }
        __syncthreads();
    }
    if (t == 0) {
        out[OUT_PPL]   = expf(sh[0]);
        out[OUT_USAGE] = su[0] * (1.0f / (float)NCODES);
    }
}

extern "C" void kernel_launch(void* const* d_in, const int* in_sizes, int n_in,
                              void* d_out, int out_size, void* d_ws, size_t ws_size,
                              hipStream_t stream) {
    const float* z_e   = (const float*)d_in[0];
    const float* embed = (const float*)d_in[1];
    float* out    = (float*)d_out;
    float* counts = (float*)d_ws;   // 512 floats of scratch

    vq_init<<<2, 256, 0, stream>>>(out, counts);

    const size_t smem_bytes = (size_t)(CODES_PER_PASS * SR + CODES_PER_PASS) * sizeof(float)
                              + 8 * 16 * sizeof(int);   // 71,168 B -> 4 WGs per WGP
    vq_main<<<NROWS / 128, 256, smem_bytes, stream>>>(z_e, embed, out, counts);

    vq_finalize<<<1, 256, 0, stream>>>(counts, out);
}